// GCN_2662879723612
// MI455X (gfx1250) — compile-verified
//
#include <hip/hip_runtime.h>
#include <hip/hip_bf16.h>

typedef float v2f __attribute__((ext_vector_type(2)));
typedef float v8f __attribute__((ext_vector_type(8)));

// ---------------------------------------------------------------------------
// Utility fills (grid-stride, deterministic)
// ---------------------------------------------------------------------------
__global__ void GCN_fill_kernel(float* __restrict__ p, float v, size_t n) {
    size_t i = blockIdx.x * (size_t)blockDim.x + threadIdx.x;
    size_t stride = (size_t)gridDim.x * blockDim.x;
    for (; i < n; i += stride) p[i] = v;
}

// ---------------------------------------------------------------------------
// Degree accumulation over destination nodes (self loop pre-counted as 1.0)
// ---------------------------------------------------------------------------
__global__ void GCN_deg_kernel(const long long* __restrict__ dst,
                               float* __restrict__ deg, int E) {
    int e = blockIdx.x * blockDim.x + threadIdx.x;
    if (e < E) atomicAdd(&deg[dst[e]], 1.0f);
}

__global__ void GCN_dinv_kernel(float* __restrict__ deg, int N) {
    int i = blockIdx.x * blockDim.x + threadIdx.x;
    if (i < N) {
        float d = deg[i];
        deg[i] = (d > 0.0f) ? rsqrtf(d) : 0.0f;
    }
}

// ---------------------------------------------------------------------------
// WMMA fp32 GEMM:  C[N x 128] = A[N x 128] * B[128 x 128]
// One block = 16 rows x 128 cols; 8 waves, each wave owns a 16x16 tile.
// K = 128 processed as 32 steps of v_wmma_f32_16x16x4_f32.
//
// Fragment layouts per CDNA5 ISA 7.12.2 (fp32):
//   A 16x4 : lanes 0-15 -> M=lane, K={0,1} in v0,v1 ; lanes 16-31 -> K={2,3}
//   B 4x16 : lanes 0-15 -> N=lane, K rows {0,1} in v0,v1 ; lanes 16-31 -> {2,3}
//   C 16x16: VGPR v, lanes 0-15 -> M=v ; lanes 16-31 -> M=v+8 ; N = lane&15
// ---------------------------------------------------------------------------
__global__ void __launch_bounds__(256, 2)
GCN_gemm_wmma_kernel(const float* __restrict__ A,
                     const float* __restrict__ B,
                     float* __restrict__ C, int Nrows) {
    const int lane = threadIdx.x & 31;
    const int wv   = threadIdx.x >> 5;   // column tile 0..7
    const int l16  = lane & 15;
    const int half = lane >> 4;          // 0 or 1
    const int row0 = blockIdx.x * 16;
    const int col0 = wv * 16;

    // Clamp the A-row so partially-full last tiles still make legal loads.
    int arow = row0 + l16;
    if (arow >= Nrows) arow = Nrows - 1;

    const float* __restrict__ aptr = A + (size_t)arow * 128 + half * 2;
    const float* __restrict__ bptr = B + col0 + l16;

    v8f acc = {};
#pragma unroll
    for (int kk = 0; kk < 32; ++kk) {
        const int k0 = kk * 4;
        v2f a = *(const v2f*)(aptr + k0);       // A[row][k0+2*half +{0,1}]
        v2f b;
        b.x = bptr[(size_t)(k0 + half * 2 + 0) * 128];
        b.y = bptr[(size_t)(k0 + half * 2 + 1) * 128];
        acc = __builtin_amdgcn_wmma_f32_16x16x4_f32(
            /*neg_a=*/false, a, /*neg_b=*/false, b,
            /*c_mod=*/(short)0, acc, /*reuse_a=*/false, /*reuse_b=*/false);
    }

#pragma unroll
    for (int v = 0; v < 8; ++v) {
        int r = row0 + half * 8 + v;
        if (r < Nrows) C[(size_t)r * 128 + col0 + l16] = acc[v];
    }
}

// ---------------------------------------------------------------------------
// Edge scatter: one wave per edge; each lane carries 4 floats of the row.
// out[dst] += hw[src] * dinv[src] * dinv[dst]
// ---------------------------------------------------------------------------
__global__ void __launch_bounds__(256)
GCN_scatter_kernel(const long long* __restrict__ srcI,
                   const long long* __restrict__ dstI,
                   const float* __restrict__ dinv,
                   const float* __restrict__ hw,
                   float* __restrict__ out, int E) {
    const int wid  = (int)((blockIdx.x * (size_t)blockDim.x + threadIdx.x) >> 5);
    const int lane = threadIdx.x & 31;
    if (wid >= E) return;

    const long long s = srcI[wid];
    const long long d = dstI[wid];
    const float norm  = dinv[s] * dinv[d];

    const float4 v = ((const float4*)(hw + (size_t)s * 128))[lane];
    float* o = out + (size_t)d * 128 + lane * 4;
    atomicAdd(o + 0, v.x * norm);
    atomicAdd(o + 1, v.y * norm);
    atomicAdd(o + 2, v.z * norm);
    atomicAdd(o + 3, v.w * norm);
}

// ---------------------------------------------------------------------------
// Fused self-loop term + bias + ReLU, in place over the scatter accumulator.
// acc[n][f] = relu(acc[n][f] + hw[n][f]*dinv[n]^2 + bias[f])
// ---------------------------------------------------------------------------
__global__ void __launch_bounds__(256)
GCN_relu_bias_self_kernel(float* __restrict__ acc,
                          const float* __restrict__ hw,
                          const float* __restrict__ dinv,
                          const float* __restrict__ bias, int N) {
    size_t i = blockIdx.x * (size_t)blockDim.x + threadIdx.x;
    if (i >= (size_t)N * 128) return;
    const int n = (int)(i >> 7);
    const int f = (int)(i & 127);
    const float di = dinv[n];
    const float v = acc[i] + hw[i] * di * di + bias[f];
    acc[i] = v > 0.0f ? v : 0.0f;
}

// ---------------------------------------------------------------------------
// Segment max pool. Values are post-ReLU (>= 0), so unsigned-bit atomicMax is
// order-preserving and a 0.0f init implements the reference's -inf -> 0 guard.
// ---------------------------------------------------------------------------
__global__ void __launch_bounds__(256)
GCN_pool_max_kernel(const float* __restrict__ h,
                    const long long* __restrict__ batch,
                    unsigned* __restrict__ pooled, int N) {
    size_t i = blockIdx.x * (size_t)blockDim.x + threadIdx.x;
    if (i >= (size_t)N * 128) return;
    const int n = (int)(i >> 7);
    const int f = (int)(i & 127);
    atomicMax(&pooled[(size_t)batch[n] * 128 + f], __float_as_uint(h[i]));
}

// ---------------------------------------------------------------------------
// Head: out[g] = [pooled[g] . W_iou + b_iou, pooled[g] . W_iop + b_iop]
// ---------------------------------------------------------------------------
__global__ void __launch_bounds__(128)
GCN_head_kernel(const float* __restrict__ pooled,
                const float* __restrict__ Wu, const float* __restrict__ bu,
                const float* __restrict__ Wp, const float* __restrict__ bp,
                float* __restrict__ out) {
    __shared__ float s1[128];
    __shared__ float s2[128];
    const int g = blockIdx.x;
    const int t = threadIdx.x;
    const float p = pooled[(size_t)g * 128 + t];
    s1[t] = p * Wu[t];
    s2[t] = p * Wp[t];
    __syncthreads();
    for (int st = 64; st > 0; st >>= 1) {
        if (t < st) { s1[t] += s1[t + st]; s2[t] += s2[t + st]; }
        __syncthreads();
    }
    if (t == 0) {
        out[g * 2 + 0] = s1[0] + bu[0];
        out[g * 2 + 1] = s2[0] + bp[0];
    }
}

// ---------------------------------------------------------------------------
extern "C" void kernel_launch(void* const* d_in, const int* in_sizes, int n_in,
                              void* d_out, int out_size, void* d_ws, size_t ws_size,
                              hipStream_t stream) {
    const float*     x     = (const float*)d_in[0];
    const long long* edge  = (const long long*)d_in[1];   // int64 [2, E]
    const long long* batch = (const long long*)d_in[2];   // int64 [N]
    const float*     W1    = (const float*)d_in[3];
    const float*     b1    = (const float*)d_in[4];
    const float*     W2    = (const float*)d_in[5];
    const float*     b2    = (const float*)d_in[6];
    const float*     Wu    = (const float*)d_in[7];
    const float*     bu    = (const float*)d_in[8];
    const float*     Wp    = (const float*)d_in[9];
    const float*     bp    = (const float*)d_in[10];

    const int N = in_sizes[0] / 128;
    const int E = in_sizes[1] / 2;
    const int G = out_size / 2;

    const long long* srcI = edge;
    const long long* dstI = edge + E;

    // Workspace carve (256B aligned regions)
    char* ws = (char*)d_ws;
    size_t off = 0;
    auto align256 = [](size_t v) { return (v + 255) & ~(size_t)255; };
    float* dinv   = (float*)(ws + off); off += align256((size_t)N * sizeof(float));
    float* pooled = (float*)(ws + off); off += align256((size_t)G * 128 * sizeof(float));
    float* buf0   = (float*)(ws + off); off += align256((size_t)N * 128 * sizeof(float));
    float* buf1   = (float*)(ws + off); off += align256((size_t)N * 128 * sizeof(float));
    (void)ws_size;

    const size_t NF = (size_t)N * 128;
    const int fillBlocks = 2048;
    const int gemmBlocks = (N + 15) / 16;
    const int edgeBlocks = (E + 7) / 8;              // 8 waves (=8 edges) per block
    const int nfBlocks   = (int)((NF + 255) / 256);

    // 1) degree (self loop = 1.0) -> dinv
    GCN_fill_kernel<<<fillBlocks, 256, 0, stream>>>(dinv, 1.0f, (size_t)N);
    GCN_deg_kernel<<<(E + 255) / 256, 256, 0, stream>>>(dstI, dinv, E);
    GCN_dinv_kernel<<<(N + 255) / 256, 256, 0, stream>>>(dinv, N);

    // 2) layer 1: buf0 = x @ W1 ; scatter -> buf1 ; fused bias/self/relu
    GCN_gemm_wmma_kernel<<<gemmBlocks, 256, 0, stream>>>(x, W1, buf0, N);
    GCN_fill_kernel<<<fillBlocks, 256, 0, stream>>>(buf1, 0.0f, NF);
    GCN_scatter_kernel<<<edgeBlocks, 256, 0, stream>>>(srcI, dstI, dinv, buf0, buf1, E);
    GCN_relu_bias_self_kernel<<<nfBlocks, 256, 0, stream>>>(buf1, buf0, dinv, b1, N);

    // 3) layer 2: buf0 = buf1 @ W2 ; scatter -> buf1 ; fused bias/self/relu
    GCN_gemm_wmma_kernel<<<gemmBlocks, 256, 0, stream>>>(buf1, W2, buf0, N);
    GCN_fill_kernel<<<fillBlocks, 256, 0, stream>>>(buf1, 0.0f, NF);
    GCN_scatter_kernel<<<edgeBlocks, 256, 0, stream>>>(srcI, dstI, dinv, buf0, buf1, E);
    GCN_relu_bias_self_kernel<<<nfBlocks, 256, 0, stream>>>(buf1, buf0, dinv, b2, N);

    // 4) pool (atomic max on non-negative float bits) + head
    GCN_fill_kernel<<<fillBlocks, 256, 0, stream>>>(pooled, 0.0f, (size_t)G * 128);
    GCN_pool_max_kernel<<<nfBlocks, 256, 0, stream>>>(buf1, batch, (unsigned*)pooled, N);
    GCN_head_kernel<<<G, 128, 0, stream>>>(pooled, Wu, bu, Wp, bp, (float*)d_out);
}